// Attention_76570676953745
// MI455X (gfx1250) — compile-verified
//
#include <hip/hip_runtime.h>
#include <math.h>

// Problem constants (from the reference): B=32, S=2048, D=P=1024
#define BB 32
#define SS 2048
#define DD 1024
#define PP 1024
#define SCH 16            // S-chunks for vbar partial sums
#define SPC (SS / SCH)    // 128 s per chunk

typedef __attribute__((ext_vector_type(2))) float v2f;
typedef __attribute__((ext_vector_type(8))) float v8f;

// ---------------------------------------------------------------------------
// C[32,1024] = A[32,1024] x B(1024x1024), fp32 via V_WMMA_F32_16X16X4_F32.
// TRANS_B=true : B element (k,n) = Bm[n*1024 + k]   (i.e. Bm stored [N,K])
// TRANS_B=false: B element (k,n) = Bm[k*1024 + n]   (i.e. Bm stored [K,N])
// One wave computes one 16x16 tile of C. 2 m-tiles x 64 n-tiles = 128 waves.
// fp32 WMMA layouts (ISA 7.12.2):
//   A 16x4 : lane L -> M = L&15 ; VGPR0/1 hold K = 2*(L>>4) + {0,1}
//   B 4x16 : lane L -> N = L&15 ; VGPR0/1 hold K = 2*(L>>4) + {0,1}
//   C 16x16: VGPR j -> M = j + 8*(L>>4), N = L&15
// ---------------------------------------------------------------------------
template <bool TRANS_B>
__global__ void gemm32_wmma(const float* __restrict__ A,
                            const float* __restrict__ Bm,
                            float* __restrict__ C) {
  const int wave  = (blockIdx.x * blockDim.x + threadIdx.x) >> 5;
  const int lane  = threadIdx.x & 31;
  const int mtile = wave >> 6;        // 0..1   (64 n-tiles)
  const int ntile = wave & 63;        // 0..63
  const int m     = lane & 15;
  const int half  = lane >> 4;        // 0 or 1
  const int kb    = half * 2;         // K sub-offset {0,2}
  const int n     = ntile * 16 + (lane & 15);

  const float* Arow = A + (size_t)(mtile * 16 + m) * DD;

  v8f acc = {};
#pragma unroll 4
  for (int k = 0; k < DD; k += 4) {
    v2f a = *(const v2f*)(Arow + k + kb);            // A[m][k+kb .. k+kb+1]
    v2f b;
    if (TRANS_B) {
      b = *(const v2f*)(Bm + (size_t)n * DD + k + kb);   // contiguous pair
    } else {
      b.x = Bm[(size_t)(k + kb)     * PP + n];
      b.y = Bm[(size_t)(k + kb + 1) * PP + n];
    }
    acc = __builtin_amdgcn_wmma_f32_16x16x4_f32(
        /*neg_a=*/false, a, /*neg_b=*/false, b,
        /*c_mod=*/(short)0, acc, /*reuse_a=*/false, /*reuse_b=*/false);
  }

  const int row0 = mtile * 16 + half * 8;
  const int col  = ntile * 16 + (lane & 15);
#pragma unroll
  for (int j = 0; j < 8; ++j)
    C[(size_t)(row0 + j) * PP + col] = acc[j];
}

// ---------------------------------------------------------------------------
// scores[b,s] = dot(k[b,s,:], t[b,:])   — streams k (256 MB), memory bound.
// grid (S/8, B), block 256 (8 waves); one wave per s row; t[b] staged in LDS.
// ---------------------------------------------------------------------------
__global__ void scores_kernel(const float* __restrict__ k,
                              const float* __restrict__ t,
                              float* __restrict__ scores) {
  __shared__ float tl[DD];
  const int b   = blockIdx.y;
  const int tid = threadIdx.x;

  ((float4*)tl)[tid] = ((const float4*)(t + (size_t)b * DD))[tid];  // 256*4=1024
  __syncthreads();

  const int wave = tid >> 5, lane = tid & 31;
  const int s = blockIdx.x * 8 + wave;
  const float* krow = k + ((size_t)b * SS + s) * DD;

  float acc = 0.f;
#pragma unroll
  for (int i = 0; i < 8; ++i) {
    const int d = lane * 4 + i * 128;
    float4 kv = *(const float4*)(krow + d);
    float4 tv = *(const float4*)(tl + d);
    acc += kv.x * tv.x + kv.y * tv.y + kv.z * tv.z + kv.w * tv.w;
  }
#pragma unroll
  for (int off = 16; off; off >>= 1) acc += __shfl_xor(acc, off, 32);
  if (lane == 0) scores[(size_t)b * SS + s] = acc;
}

// ---------------------------------------------------------------------------
// Masked softmax over s, one block (1024 threads) per b; writes E to d_out.
// ---------------------------------------------------------------------------
__global__ void softmax_kernel(const float* __restrict__ scores,
                               const int* __restrict__ k_lens,
                               float* __restrict__ E) {
  __shared__ float redm[32], reds[32];
  __shared__ float bmax, bsum;
  const int b = blockIdx.x, tid = threadIdx.x;
  const int len = k_lens[b];

  float s0 = (tid        < len) ? scores[(size_t)b * SS + tid       ] : -INFINITY;
  float s1 = (tid + 1024 < len) ? scores[(size_t)b * SS + tid + 1024] : -INFINITY;

  float m = fmaxf(s0, s1);
#pragma unroll
  for (int o = 16; o; o >>= 1) m = fmaxf(m, __shfl_xor(m, o, 32));
  if ((tid & 31) == 0) redm[tid >> 5] = m;
  __syncthreads();
  if (tid < 32) {
    float mm = redm[tid];
#pragma unroll
    for (int o = 16; o; o >>= 1) mm = fmaxf(mm, __shfl_xor(mm, o, 32));
    if (tid == 0) bmax = mm;
  }
  __syncthreads();
  m = bmax;

  float e0 = __expf(s0 - m);     // exp(-inf) -> 0 for masked lanes
  float e1 = __expf(s1 - m);
  float su = e0 + e1;
#pragma unroll
  for (int o = 16; o; o >>= 1) su += __shfl_xor(su, o, 32);
  if ((tid & 31) == 0) reds[tid >> 5] = su;
  __syncthreads();
  if (tid < 32) {
    float ss = reds[tid];
#pragma unroll
    for (int o = 16; o; o >>= 1) ss += __shfl_xor(ss, o, 32);
    if (tid == 0) bsum = ss;
  }
  __syncthreads();

  const float inv = 1.0f / bsum;
  E[(size_t)b * SS + tid       ] = e0 * inv;
  E[(size_t)b * SS + tid + 1024] = e1 * inv;
}

// ---------------------------------------------------------------------------
// vbar partials: vpart[b,c,d] = sum_{s in chunk c} E[b,s] * v[b,s,d]
// Streams v (256 MB). grid (SCH, B), block 256; thread owns 4 d's (float4).
// Deterministic (no float atomics) -> fixed-order tree reduce afterwards.
// ---------------------------------------------------------------------------
__global__ void vbar_part_kernel(const float* __restrict__ v,
                                 const float* __restrict__ E,
                                 float* __restrict__ vpart) {
  __shared__ float el[SPC];
  const int b = blockIdx.y, c = blockIdx.x, tid = threadIdx.x;
  const int s0 = c * SPC;
  if (tid < SPC) el[tid] = E[(size_t)b * SS + s0 + tid];
  __syncthreads();

  const int d = tid * 4;
  float4 acc = {0.f, 0.f, 0.f, 0.f};
  const float* vb = v + ((size_t)b * SS + s0) * DD + d;
  for (int s = 0; s < SPC; ++s) {
    float4 vv = *(const float4*)(vb + (size_t)s * DD);
    const float e = el[s];
    acc.x += e * vv.x; acc.y += e * vv.y;
    acc.z += e * vv.z; acc.w += e * vv.w;
  }
  *(float4*)(vpart + ((size_t)(b * SCH + c)) * DD + d) = acc;
}

__global__ void vbar_reduce_kernel(const float* __restrict__ vpart,
                                   float* __restrict__ vbar) {
  const int b = blockIdx.y;
  const int d = blockIdx.x * 256 + threadIdx.x;
  float acc = 0.f;
#pragma unroll
  for (int c = 0; c < SCH; ++c)
    acc += vpart[((size_t)(b * SCH + c)) * DD + d];
  vbar[(size_t)b * DD + d] = acc;
}

// ---------------------------------------------------------------------------
extern "C" void kernel_launch(void* const* d_in, const int* in_sizes, int n_in,
                              void* d_out, int out_size, void* d_ws, size_t ws_size,
                              hipStream_t stream) {
  const float* k      = (const float*)d_in[0];  // [B,S,D]
  const float* q      = (const float*)d_in[1];  // [B,D]
  const float* v      = (const float*)d_in[2];  // [B,S,D]
  const int*   k_lens = (const int*)  d_in[3];  // [B]
  const float* Wk     = (const float*)d_in[4];  // [P,D]

  float* attended = (float*)d_out;              // [B,P]  (first output)
  float* E        = attended + BB * PP;         // [B,S,1] (second output)

  float* ws     = (float*)d_ws;                 // ~2.7 MB used
  float* enc_q  = ws;                           // [B,P]     32K floats
  float* t      = enc_q  + BB * PP;             // [B,D]     32K floats
  float* scores = t      + BB * DD;             // [B,S]     64K floats
  float* vpart  = scores + BB * SS;             // [B,SCH,D] 512K floats
  float* vbar   = vpart  + BB * SCH * DD;       // [B,D]     32K floats

  // enc_q = q @ Wk^T         (B[k=d][n=p] = Wk[p*D+d] -> transposed access)
  gemm32_wmma<true ><<<32, 128, 0, stream>>>(q, Wk, enc_q);
  // t = enc_q @ Wk           (B[k=p][n=d] = Wk[p*D+d] -> natural access)
  gemm32_wmma<false><<<32, 128, 0, stream>>>(enc_q, Wk, t);
  // scores[b,s] = k[b,s,:] . t[b,:]
  scores_kernel<<<dim3(SS / 8, BB), 256, 0, stream>>>(k, t, scores);
  // E = masked softmax(scores)  (written straight into d_out)
  softmax_kernel<<<BB, 1024, 0, stream>>>(scores, k_lens, E);
  // vbar[b,:] = sum_s E[b,s] * v[b,s,:]
  vbar_part_kernel<<<dim3(SCH, BB), 256, 0, stream>>>(v, E, vpart);
  vbar_reduce_kernel<<<dim3(DD / 256, BB), 256, 0, stream>>>(vpart, vbar);
  // attended = vbar @ Wk^T   (transposed access again)
  gemm32_wmma<true ><<<32, 128, 0, stream>>>(vbar, Wk, attended);
}